// ImageModel_74895639707992
// MI455X (gfx1250) — compile-verified
//
#include <hip/hip_runtime.h>

// ---------------------------------------------------------------------------
// Gaussian peak scatter for MI455X (gfx1250, wave32).
//
// 50,000 Gaussian peaks (25x25 windows) scatter-added into a 2048x2048 f32
// image + background.  Output (16.8 MB) lives in the 192 MB L2; the bound is
// ~31M global_atomic_add_f32 resolved in L2 (~125 MB RMW).
//
// Separable Gaussian => each peak window is a rank-1 outer product
// val[i][j] = h*ey[i]*ex[j], computed as 2x2 tiles of 16x16 with
// V_WMMA_F32_16X16X4_F32 (K=1), one wave32 per peak.  Scatter uses an
// interior fast path (97.7% of peaks): a single 64-bit base address per peak
// and compile-time instruction offsets for all 32 atomics, with only ~4
// EXEC-mask regions (the lane-structural i<25 / j<25 masks).
// ---------------------------------------------------------------------------

typedef __attribute__((ext_vector_type(2))) float v2f;
typedef __attribute__((ext_vector_type(8))) float v8f;

#define IMG_H 2048
#define IMG_W 2048
#define WIN 12           // window half-size; window is 25x25
#define WSIZE 25

// ---- Kernel 1: initialize image to background (d_out is poisoned). --------
__global__ __launch_bounds__(256) void img_init_kernel(float* __restrict__ out,
                                                       const float* __restrict__ bg,
                                                       int n4) {
    const float b = bg[0];
    const int idx = blockIdx.x * blockDim.x + threadIdx.x;
    if (idx < n4) {
        float4 v = make_float4(b, b, b, b);
        *(float4*)(out + 4 * idx) = v;   // out_size (4M) is divisible by 4
    }
}

// ---- Kernel 2: one wave32 per peak; 4 WMMA outer-product tiles + scatter. -
__global__ __launch_bounds__(256) void peaks_scatter_kernel(
    const float* __restrict__ px, const float* __restrict__ py,
    const float* __restrict__ ph, const float* __restrict__ pw,
    float* __restrict__ out, int npeaks) {

    const int lane = threadIdx.x & 31;
    const int wave = threadIdx.x >> 5;
    const int peak = blockIdx.x * (blockDim.x >> 5) + wave;
    if (peak >= npeaks) return;          // wave-uniform: EXEC stays all-1s

    const float x = px[peak], y = py[peak];
    const float h = ph[peak], wd = pw[peak];
    const float bx = floorf(x), by = floorf(y);
    const float fx = x - bx,   fy = y - by;      // fractional position in pixel
    const float inv2w2 = 1.0f / (2.0f * wd * wd);
    const int ibx = (int)bx, iby = (int)by;

    // 1-D Gaussian factors.  m = lane%16; group 0 covers window index 0..15,
    // group 1 covers 16..24 (rest zeroed).
    const int m = lane & 15;
    const float d0y = (float)(m - WIN) - fy;
    const float d1y = (float)(m + 16 - WIN) - fy;
    const float d0x = (float)(m - WIN) - fx;
    const float d1x = (float)(m + 16 - WIN) - fx;
    const float ey0 = __expf(-d0y * d0y * inv2w2);
    const float ey1 = (m + 16 < WSIZE) ? __expf(-d1y * d1y * inv2w2) : 0.0f;
    const float ex0 = h * __expf(-d0x * d0x * inv2w2);
    const float ex1 = (m + 16 < WSIZE) ? (h * __expf(-d1x * d1x * inv2w2)) : 0.0f;

    // WMMA operand layouts (16x16x4 f32, wave32):
    //   A (16x4): lanes 0-15 give M=0..15; VGPR0 = K=0 -> ey in .x; rest 0.
    //   B (4x16): VGPR0, lanes 0-15 hold K=0,N=lane -> h*ex in .x; rest 0.
    const bool lo = (lane < 16);
    v2f a0; a0.x = lo ? ey0 : 0.0f; a0.y = 0.0f;
    v2f a1; a1.x = lo ? ey1 : 0.0f; a1.y = 0.0f;
    v2f b0; b0.x = lo ? ex0 : 0.0f; b0.y = 0.0f;
    v2f b1; b1.x = lo ? ex1 : 0.0f; b1.y = 0.0f;
    v8f czero = {};

    // Four independent back-to-back WMMAs (no D->A/B hazards between them).
    // C/D layout: lane L, element v  ->  M = v + 8*(L/16), N = L%16.
    v8f c00 = __builtin_amdgcn_wmma_f32_16x16x4_f32(false, a0, false, b0,
                                                    (short)0, czero, false, false);
    v8f c01 = __builtin_amdgcn_wmma_f32_16x16x4_f32(false, a0, false, b1,
                                                    (short)0, czero, false, false);
    v8f c10 = __builtin_amdgcn_wmma_f32_16x16x4_f32(false, a1, false, b0,
                                                    (short)0, czero, false, false);
    v8f c11 = __builtin_amdgcn_wmma_f32_16x16x4_f32(false, a1, false, b1,
                                                    (short)0, czero, false, false);

    const int hi8 = (lane >> 4) * 8;     // 0 for lanes 0-15, 8 for lanes 16-31
    const int n = lane & 15;             // column within tile

    // Interior test: whole 25x25 window inside the image (true for ~97.7% of
    // peaks; lanes of a wave always agree, so the branch is one region).
    const bool interior = (ibx >= WIN) & (ibx < IMG_W - WIN) &
                          (iby >= WIN) & (iby < IMG_H - WIN);

    if (interior) {
        // Single base address; every atomic below uses a constant byte offset
        // (row stride 8192 B, max offset 23*8192+16*4 < 2^23 -> fits IOFFSET).
        float* p = out + (size_t)(iby - WIN + hi8) * IMG_W + (ibx - WIN + n);

        // Tile (0,0): rows hi8+0..7 (0..15), cols n (0..15) -> all valid.
        #pragma unroll
        for (int v = 0; v < 8; ++v)
            atomicAdd(p + v * IMG_W, c00[v]);

        // Tile (0,1): cols 16+n, valid iff n <= 8.
        if (n <= 8) {
            #pragma unroll
            for (int v = 0; v < 8; ++v)
                atomicAdd(p + v * IMG_W + 16, c01[v]);
        }

        // Tile (1,0): rows 16+hi8+v, valid iff hi8+v <= 8.
        atomicAdd(p + 16 * IMG_W, c10[0]);            // v=0: row 16+hi8 <= 24
        if (hi8 == 0) {
            #pragma unroll
            for (int v = 1; v < 8; ++v)
                atomicAdd(p + (16 + v) * IMG_W, c10[v]);
        }

        // Tile (1,1): rows 16+hi8+v AND cols 16+n.
        if (n <= 8) {
            atomicAdd(p + 16 * IMG_W + 16, c11[0]);   // v=0
            if (hi8 == 0) {
                #pragma unroll
                for (int v = 1; v < 8; ++v)
                    atomicAdd(p + (16 + v) * IMG_W + 16, c11[v]);
            }
        }
    } else {
        // Border peak (~2.3%): fully masked scatter, matches reference mask.
        #define SCATTER_GEN(C, TI, TJ)                                        \
            {                                                                 \
                const int j  = (TJ) * 16 + n;                                 \
                const int gx = ibx - WIN + j;                                 \
                _Pragma("unroll")                                             \
                for (int v = 0; v < 8; ++v) {                                 \
                    const int i  = (TI) * 16 + hi8 + v;                       \
                    const int gy = iby - WIN + i;                             \
                    if (i < WSIZE && j < WSIZE &&                             \
                        gx >= 0 && gx < IMG_W && gy >= 0 && gy < IMG_H) {     \
                        atomicAdd(out + (size_t)gy * IMG_W + gx, (C)[v]);     \
                    }                                                         \
                }                                                             \
            }
        SCATTER_GEN(c00, 0, 0)
        SCATTER_GEN(c01, 0, 1)
        SCATTER_GEN(c10, 1, 0)
        SCATTER_GEN(c11, 1, 1)
        #undef SCATTER_GEN
    }
}

extern "C" void kernel_launch(void* const* d_in, const int* in_sizes, int n_in,
                              void* d_out, int out_size, void* d_ws, size_t ws_size,
                              hipStream_t stream) {
    // setup_inputs order: pos_x, pos_y, height, width, background, x_grid, y_grid
    const float* px = (const float*)d_in[0];
    const float* py = (const float*)d_in[1];
    const float* ph = (const float*)d_in[2];
    const float* pw = (const float*)d_in[3];
    const float* bg = (const float*)d_in[4];
    float* out = (float*)d_out;

    const int npeaks = in_sizes[0];

    // 1) out = background everywhere (float4 stores, stays resident in L2)
    const int n4 = out_size / 4;
    img_init_kernel<<<(n4 + 255) / 256, 256, 0, stream>>>(out, bg, n4);

    // 2) one wave32 per peak, 8 waves (256 threads) per block
    const int waves_per_block = 8;
    const int blocks = (npeaks + waves_per_block - 1) / waves_per_block;
    peaks_scatter_kernel<<<blocks, 256, 0, stream>>>(px, py, ph, pw, out, npeaks);
}